// CRF_5772436046015
// MI455X (gfx1250) — compile-verified
//
#include <hip/hip_runtime.h>
#include <hip/hip_bf16.h>
#include <stdint.h>

#define CRF_T 48
#define CRF_S 1024
#define CRF_B 512

typedef float v2f __attribute__((ext_vector_type(2)));
typedef float v8f __attribute__((ext_vector_type(8)));

// ds_swizzle group-of-32: imm = {1'b0, xor[14:10], or[9:5]=0, and[4:0]=0x1F}
#define SWZF(x, m) __int_as_float(__builtin_amdgcn_ds_swizzle(__float_as_int(x), ((m) << 10) | 0x1F))
#define SWZI(x, m) __builtin_amdgcn_ds_swizzle((int)(x), ((m) << 10) | 0x1F)

// Forward (log-partition) kernel.
// One wave (32 threads) handles 16 consecutive batches as the M-rows of
// V_WMMA_F32_16X16X4_F32. Layouts:
//   C/D (score): vgpr m <-> batch (m + 8*half), lane (ln + 16*tile) <-> tag.
//   A (p = exp(score - max)): lane ln <-> batch, K = 4*kc + 2*half + v.
//   B (E^T): value at (K=i, N=j) = exp(trans[j*48 + i])   (reference uses trans[next, prev]).
__global__ __launch_bounds__(32) void crf_forward_wmma(
    const float* __restrict__ emissions,
    const uint8_t* __restrict__ mask,
    const float* __restrict__ start_t,
    const float* __restrict__ end_t,
    const float* __restrict__ trans,
    float* __restrict__ denom)
{
    __shared__ float pbuf[16 * CRF_T];

    const int lane = threadIdx.x & 31;
    const int ln = lane & 15;
    const int hf = lane >> 4;
    const int b0 = blockIdx.x * 16;

    // ---- precompute B-matrix fragments: exp(transitions) transposed ----
    v2f Bf[12][3];
#pragma unroll
    for (int kc = 0; kc < 12; ++kc) {
#pragma unroll
        for (int tl = 0; tl < 3; ++tl) {
            const int i0 = 4 * kc + 2 * hf;
            const int j = ln + 16 * tl;
            v2f e;
            e.x = __expf(trans[j * CRF_T + i0]);
            e.y = __expf(trans[j * CRF_T + i0 + 1]);
            Bf[kc][tl] = e;
        }
    }

    // ---- sequence lengths for this lane-half's batches (sum of mask bytes) ----
    int lenv[8];
#pragma unroll
    for (int m = 0; m < 8; ++m) {
        const int b = b0 + m + 8 * hf;
        const uint32_t* mrow = (const uint32_t*)(mask + (size_t)b * CRF_S);
        uint32_t cnt = 0;
#pragma unroll
        for (int k = 0; k < 16; ++k) {
            const uint32_t x = mrow[ln + 16 * k];
            cnt += (x * 0x01010101u) >> 24;   // sum of 4 bytes (each 0/1)
        }
        cnt += (uint32_t)SWZI(cnt, 1);
        cnt += (uint32_t)SWZI(cnt, 2);
        cnt += (uint32_t)SWZI(cnt, 4);
        cnt += (uint32_t)SWZI(cnt, 8);
        lenv[m] = (int)cnt;
    }

    // ---- initial score: start + emissions[:, 0, :] (C/D layout) ----
    v8f sc[3];
#pragma unroll
    for (int tl = 0; tl < 3; ++tl) {
        const int j = ln + 16 * tl;
        const float st = start_t[j];
        v8f c;
#pragma unroll
        for (int m = 0; m < 8; ++m) {
            const int b = b0 + m + 8 * hf;
            c[m] = st + emissions[(size_t)b * CRF_S * CRF_T + j];
        }
        sc[tl] = c;
    }

    float* const pst = &pbuf[hf * 8 * CRF_T + ln];               // store base
    const v2f* const pld = (const v2f*)&pbuf[ln * CRF_T + 2 * hf]; // A-frag load base

    for (int t = 1; t < CRF_S; ++t) {
        // per-batch max over the 48 tags (3-way max + xor butterfly in 16-lane half)
        float mb[8];
#pragma unroll
        for (int m = 0; m < 8; ++m) {
            float mx = fmaxf(fmaxf(sc[0][m], sc[1][m]), sc[2][m]);
            mx = fmaxf(mx, SWZF(mx, 1));
            mx = fmaxf(mx, SWZF(mx, 2));
            mx = fmaxf(mx, SWZF(mx, 4));
            mx = fmaxf(mx, SWZF(mx, 8));
            mb[m] = mx;
        }

        // p = exp(score - max) -> LDS in [batch-local][tag] order
#pragma unroll
        for (int tl = 0; tl < 3; ++tl)
#pragma unroll
            for (int m = 0; m < 8; ++m)
                pst[m * CRF_T + 16 * tl] = __expf(sc[tl][m] - mb[m]);
        asm volatile("s_wait_dscnt 0" ::: "memory");

        // 16x48 @ 48x48 in the linear domain: 12 K-chunks x 3 N-tiles
        v8f acc[3] = {};
#pragma unroll
        for (int kc = 0; kc < 12; ++kc) {
            const v2f a = pld[2 * kc];
#pragma unroll
            for (int tl = 0; tl < 3; ++tl) {
                acc[tl] = __builtin_amdgcn_wmma_f32_16x16x4_f32(
                    false, a, false, Bf[kc][tl], (short)0, acc[tl], false, false);
            }
        }

        // score' = max + log(acc) + emission, gated by mask (t < len)
#pragma unroll
        for (int tl = 0; tl < 3; ++tl) {
            v8f s = sc[tl];
            const v8f c = acc[tl];
#pragma unroll
            for (int m = 0; m < 8; ++m) {
                const int b = b0 + m + 8 * hf;
                const float em =
                    emissions[(size_t)b * CRF_S * CRF_T + (size_t)t * CRF_T + ln + 16 * tl];
                const float ns = mb[m] + __logf(c[m]) + em;
                s[m] = (t < lenv[m]) ? ns : s[m];
            }
            sc[tl] = s;
        }
    }

    // ---- tail: denom[b] = logsumexp_j(score[b][j] + end[j]) ----
    float ev[3];
#pragma unroll
    for (int tl = 0; tl < 3; ++tl) ev[tl] = end_t[ln + 16 * tl];
#pragma unroll
    for (int m = 0; m < 8; ++m) {
        const float v0 = sc[0][m] + ev[0];
        const float v1 = sc[1][m] + ev[1];
        const float v2 = sc[2][m] + ev[2];
        float mx = fmaxf(fmaxf(v0, v1), v2);
        mx = fmaxf(mx, SWZF(mx, 1));
        mx = fmaxf(mx, SWZF(mx, 2));
        mx = fmaxf(mx, SWZF(mx, 4));
        mx = fmaxf(mx, SWZF(mx, 8));
        float s = __expf(v0 - mx) + __expf(v1 - mx) + __expf(v2 - mx);
        s += SWZF(s, 1);
        s += SWZF(s, 2);
        s += SWZF(s, 4);
        s += SWZF(s, 8);
        if (ln == 0) denom[b0 + m + 8 * hf] = mx + __logf(s);
    }
}

// Numerator: per-batch gather along the gold path. One 64-thread block per batch.
__global__ __launch_bounds__(64) void crf_numer(
    const float* __restrict__ emissions,
    const int* __restrict__ tags,
    const uint8_t* __restrict__ mask,
    const float* __restrict__ start_t,
    const float* __restrict__ end_t,
    const float* __restrict__ trans,
    float* __restrict__ numer)
{
    __shared__ float ssum[64];
    __shared__ int scnt[64];
    const int b = blockIdx.x;
    const int tid = threadIdx.x;
    const int* trow = tags + (size_t)b * CRF_S;
    const uint8_t* mrow = mask + (size_t)b * CRF_S;
    const float* erow = emissions + (size_t)b * CRF_S * CRF_T;

    float sum = 0.0f;
    int cnt = 0;
    for (int t = tid; t < CRF_S; t += 64) {
        const int mk = mrow[t];
        cnt += mk;
        if (t >= 1 && mk) {
            const int tp = trow[t - 1];
            const int tc = trow[t];
            sum += trans[tp * CRF_T + tc] + erow[(size_t)t * CRF_T + tc];
        }
    }
    ssum[tid] = sum;
    scnt[tid] = cnt;
    __syncthreads();
#pragma unroll
    for (int off = 32; off > 0; off >>= 1) {
        if (tid < off) {
            ssum[tid] += ssum[tid + off];
            scnt[tid] += scnt[tid + off];
        }
        __syncthreads();
    }
    if (tid == 0) {
        const int t0 = trow[0];
        const int last = scnt[0] - 1;
        numer[b] = start_t[t0] + erow[t0] + ssum[0] + end_t[trow[last]];
    }
}

// mean(denom - numer) over 512 batches -> scalar f32
__global__ __launch_bounds__(256) void crf_mean(
    const float* __restrict__ denom,
    const float* __restrict__ numer,
    float* __restrict__ out)
{
    __shared__ float s[256];
    const int tid = threadIdx.x;
    float v = (denom[tid] - numer[tid]) + (denom[tid + 256] - numer[tid + 256]);
    s[tid] = v;
    __syncthreads();
#pragma unroll
    for (int off = 128; off > 0; off >>= 1) {
        if (tid < off) s[tid] += s[tid + off];
        __syncthreads();
    }
    if (tid == 0) out[0] = s[0] * (1.0f / (float)CRF_B);
}

extern "C" void kernel_launch(void* const* d_in, const int* in_sizes, int n_in,
                              void* d_out, int out_size, void* d_ws, size_t ws_size,
                              hipStream_t stream) {
    (void)in_sizes; (void)n_in; (void)out_size; (void)ws_size;
    const float*   emissions = (const float*)d_in[0];
    const int*     tags      = (const int*)d_in[1];
    const uint8_t* mask      = (const uint8_t*)d_in[2];
    const float*   start_t   = (const float*)d_in[3];
    const float*   end_t     = (const float*)d_in[4];
    const float*   trans     = (const float*)d_in[5];

    float* denom = (float*)d_ws;           // 512 floats
    float* numer = denom + CRF_B;          // 512 floats

    crf_forward_wmma<<<CRF_B / 16, 32, 0, stream>>>(emissions, mask, start_t, end_t, trans, denom);
    crf_numer<<<CRF_B, 64, 0, stream>>>(emissions, tags, mask, start_t, end_t, trans, numer);
    crf_mean<<<1, 256, 0, stream>>>(denom, numer, (float*)d_out);
}